// ContextMatcher_53575422050442
// MI455X (gfx1250) — compile-verified
//
#include <hip/hip_runtime.h>
#include <hip/hip_bf16.h>
#include <math.h>
#include <float.h>

#define V     20000
#define D     256
#define NB    64
#define TOPK  50
#define MTILES (V / 16)   // 1250 column tiles
#define MROWS 32          // rows per workgroup in K2 (2 M-tiles -> 2x B reuse)
#define MBLKS (V / MROWS) // 625

typedef float v2f __attribute__((ext_vector_type(2)));
typedef float v8f __attribute__((ext_vector_type(8)));

// Wave-local LDS producer/consumer sync: lanes of one wave exchange through a
// wave-private LDS tile; hardware keeps same-wave DS ops in order, these stop
// the compiler from reordering around the exchange.
__device__ __forceinline__ void wave_sync() {
    __builtin_amdgcn_fence(__ATOMIC_RELEASE, "wavefront");
    __builtin_amdgcn_wave_barrier();
    __builtin_amdgcn_fence(__ATOMIC_ACQUIRE, "wavefront");
}

// ---------------------------------------------------------------- K1: normalize
__global__ __launch_bounds__(256) void norm_kernel(const float* __restrict__ emb,
                                                   float* __restrict__ nemb) {
    const int row  = blockIdx.x * 8 + (threadIdx.x >> 5);
    const int lane = threadIdx.x & 31;
    const float* p = emb + (size_t)row * D;
    float vals[8];
    float ss = 0.f;
#pragma unroll
    for (int i = 0; i < 8; ++i) {
        vals[i] = p[lane + 32 * i];
        ss += vals[i] * vals[i];
    }
#pragma unroll
    for (int m = 16; m > 0; m >>= 1) ss += __shfl_xor(ss, m, 32);
    const float scale = 1.0f / fmaxf(sqrtf(ss), 1e-12f);
    float* o = nemb + (size_t)row * D;
#pragma unroll
    for (int i = 0; i < 8; ++i) o[lane + 32 * i] = vals[i] * scale;
}

// ------------------------------------------- K2: blocked cosine GEMM + top-50
// 32-row stripe per workgroup: each streamed B tile feeds TWO wmma's.
__global__ __launch_bounds__(256) void cmap_kernel(const float* __restrict__ nemb,
                                                   float* __restrict__ out_cmap,
                                                   int* __restrict__ cmap_i) {
    __shared__ float As[MROWS][260];          // A stripe, padded (bank-conflict free)
    __shared__ float tile[8][MROWS][17];      // per-wave C-tile scratch (2 M-tiles)
    __shared__ float lsc[8][MROWS][TOPK];     // per-wave per-row ascending top-50
    __shared__ int   lco[8][MROWS][TOPK];

    const int tid  = threadIdx.x;
    const int w    = tid >> 5;
    const int lane = tid & 31;
    const int lr   = lane & 15;
    const int koff = (lane >> 4) << 1;        // 0 or 2 (k-slot per half-wave)
    const int m0   = blockIdx.x * MROWS;

    for (int i = tid; i < MROWS * D; i += 256) {
        const int r = i >> 8, c = i & 255;
        As[r][c] = nemb[(size_t)(m0 + r) * D + c];
    }
    __syncthreads();

    int    cnt0 = 0, cnt1 = 0;                // meaningful on lanes 0..15
    float* mys0 = &lsc[w][lr][0];
    int*   myc0 = &lco[w][lr][0];
    float* mys1 = &lsc[w][16 + lr][0];
    int*   myc1 = &lco[w][16 + lr][0];

    for (int t = w; t < MTILES; t += 8) {
        const int n0 = t * 16;
        const float* bp = nemb + (size_t)(n0 + lr) * D + koff;
        if (t + 8 < MTILES)
            __builtin_prefetch(nemb + (size_t)(n0 + 128 + lr) * D, 0, 1);

        v8f acc0 = {};
        v8f acc1 = {};
#pragma unroll 8
        for (int k = 0; k < D; k += 4) {
            v2f b = *(const v2f*)(bp + k);                 // 1 global load ...
            v2f a0; a0.x = As[lr][k + koff];      a0.y = As[lr][k + koff + 1];
            v2f a1; a1.x = As[16 + lr][k + koff]; a1.y = As[16 + lr][k + koff + 1];
            acc0 = __builtin_amdgcn_wmma_f32_16x16x4_f32(false, a0, false, b,
                                                         (short)0, acc0, false, false);
            acc1 = __builtin_amdgcn_wmma_f32_16x16x4_f32(false, a1, false, b,
                                                         (short)0, acc1, false, false);
        }
        // C layout: VGPR v, lanes 0-15 -> M=v, lanes 16-31 -> M=v+8; N = lane%16
        const int mh = (lane >> 4) << 3;
#pragma unroll
        for (int vv = 0; vv < 8; ++vv) tile[w][mh + vv][lr]      = acc0[vv];
#pragma unroll
        for (int vv = 0; vv < 8; ++vv) tile[w][16 + mh + vv][lr] = acc1[vv];
        wave_sync();

        if (lane < 16) {                      // lane owns rows lr and lr+16
            for (int j = 0; j < 16; ++j) {
                const float s  = tile[w][lr][j];
                const int  col = n0 + j;
                if (cnt0 < TOPK) {
                    int idx = cnt0;
                    while (idx > 0 && mys0[idx - 1] >= s) {
                        mys0[idx] = mys0[idx - 1]; myc0[idx] = myc0[idx - 1]; --idx;
                    }
                    mys0[idx] = s; myc0[idx] = col; ++cnt0;
                } else if (s > mys0[0]) {
                    int idx = 1;
                    while (idx < TOPK && mys0[idx] < s) {
                        mys0[idx - 1] = mys0[idx]; myc0[idx - 1] = myc0[idx]; ++idx;
                    }
                    mys0[idx - 1] = s; myc0[idx - 1] = col;
                }
            }
            for (int j = 0; j < 16; ++j) {
                const float s  = tile[w][16 + lr][j];
                const int  col = n0 + j;
                if (cnt1 < TOPK) {
                    int idx = cnt1;
                    while (idx > 0 && mys1[idx - 1] >= s) {
                        mys1[idx] = mys1[idx - 1]; myc1[idx] = myc1[idx - 1]; --idx;
                    }
                    mys1[idx] = s; myc1[idx] = col; ++cnt1;
                } else if (s > mys1[0]) {
                    int idx = 1;
                    while (idx < TOPK && mys1[idx] < s) {
                        mys1[idx - 1] = mys1[idx]; myc1[idx - 1] = myc1[idx]; ++idx;
                    }
                    mys1[idx - 1] = s; myc1[idx - 1] = col;
                }
            }
        }
        wave_sync();                          // WAR: tile reused next iteration
    }
    __syncthreads();

    // 8-way tail merge of ascending lists -> descending top-50 (ties: lower col)
    // 32 rows, 8 waves -> 4 rows per wave, lanes 0..3
    if (lane < 4) {
        const int r = (w << 2) + lane;
        int ptr[8];
#pragma unroll
        for (int i = 0; i < 8; ++i) ptr[i] = TOPK - 1;
        float* co = out_cmap + (size_t)(m0 + r) * TOPK;
        int*   ci = cmap_i  + (size_t)(m0 + r) * TOPK;
        for (int k = 0; k < TOPK; ++k) {
            float bs = -FLT_MAX; int bc = 0x7fffffff; int bw = 0;
#pragma unroll
            for (int wv = 0; wv < 8; ++wv) {
                const int   p = ptr[wv];
                const float s = lsc[wv][r][p];
                const int   c = lco[wv][r][p];
                if (s > bs || (s == bs && c < bc)) { bs = s; bc = c; bw = wv; }
            }
            ptr[bw]--;
            co[k] = (float)bc;
            ci[k] = bc;
        }
    }
}

// -------------------------------- K3: pool gather + bitonic sort + unique(pad 0)
__global__ __launch_bounds__(1024) void cand_kernel(const int* __restrict__ x,
                                                    const int* __restrict__ Kp,
                                                    const int* __restrict__ cmap_i,
                                                    float* __restrict__ out_cand,
                                                    int* __restrict__ cand_i) {
    __shared__ int sv[1024];
    __shared__ int sc[1024];
    __shared__ int outv[1024];
    const int tid = threadIdx.x;
    const int K   = Kp[0];
    const int C   = NB * K;

    int val = 0x7fffffff;                     // sentinel sorts to the end
    if (tid < C) {
        const int b = tid / K, k = tid - b * K;
        val = cmap_i[(size_t)x[b] * TOPK + k];
    }
    sv[tid] = val;
    __syncthreads();

    for (int ksz = 2; ksz <= 1024; ksz <<= 1) {
        for (int j = ksz >> 1; j > 0; j >>= 1) {
            const int partner = tid ^ j;
            if (partner > tid) {
                const int a = sv[tid], b = sv[partner];
                const bool asc = (tid & ksz) == 0;
                if (asc ? (a > b) : (a < b)) { sv[tid] = b; sv[partner] = a; }
            }
            __syncthreads();
        }
    }

    const int myv  = sv[tid];
    const int flag = (myv != 0x7fffffff) && (tid == 0 || sv[tid - 1] != myv);
    sc[tid] = flag;
    __syncthreads();
    for (int off = 1; off < 1024; off <<= 1) {
        const int add = (tid >= off) ? sc[tid - off] : 0;
        __syncthreads();
        sc[tid] += add;
        __syncthreads();
    }
    outv[tid] = 0;                            // fill_value = 0
    __syncthreads();
    if (flag) outv[sc[tid] - 1] = myv;
    __syncthreads();
    if (tid < C) {
        out_cand[tid] = (float)outv[tid];
        cand_i[tid]   = outv[tid];
    }
}

// -------------------------------- K4: V x C GEMM + per-row argmax / max
__global__ __launch_bounds__(256) void voronoi_kernel(const float* __restrict__ nemb,
                                                      const int* __restrict__ cand_i,
                                                      const int* __restrict__ Kp,
                                                      float* __restrict__ d_out) {
    __shared__ float As[16][260];
    __shared__ float tile[8][16][17];
    __shared__ float redS[8][16];
    __shared__ int   redC[8][16];

    const int tid  = threadIdx.x;
    const int w    = tid >> 5;
    const int lane = tid & 31;
    const int lr   = lane & 15;
    const int koff = (lane >> 4) << 1;
    const int m0   = blockIdx.x * 16;
    const int K    = Kp[0];
    const int C    = NB * K;
    const int tiles = C >> 4;

    for (int i = tid; i < 16 * D; i += 256) {
        const int r = i >> 8, c = i & 255;
        As[r][c] = nemb[(size_t)(m0 + r) * D + c];
    }
    __syncthreads();

    float bestS = -FLT_MAX;
    int   bestC = 0x7fffffff;
    for (int t = w; t < tiles; t += 8) {
        const int n0   = t * 16;
        const int brow = cand_i[n0 + lr];
        const float* bp = nemb + (size_t)brow * D + koff;
        v8f acc = {};
#pragma unroll 8
        for (int k = 0; k < D; k += 4) {
            v2f a; a.x = As[lr][k + koff]; a.y = As[lr][k + koff + 1];
            v2f b = *(const v2f*)(bp + k);
            acc = __builtin_amdgcn_wmma_f32_16x16x4_f32(false, a, false, b,
                                                        (short)0, acc, false, false);
        }
        const int mh = (lane >> 4) << 3;
#pragma unroll
        for (int vv = 0; vv < 8; ++vv) tile[w][mh + vv][lr] = acc[vv];
        wave_sync();
        if (lane < 16) {
            for (int j = 0; j < 16; ++j) {
                const float s = tile[w][lr][j];
                const int   c = n0 + j;
                if (s > bestS || (s == bestS && c < bestC)) { bestS = s; bestC = c; }
            }
        }
        wave_sync();
    }
    if (lane < 16) { redS[w][lr] = bestS; redC[w][lr] = bestC; }
    __syncthreads();

    if (tid < 16) {
        float bs = -FLT_MAX; int bc = 0x7fffffff;
#pragma unroll
        for (int wv = 0; wv < 8; ++wv) {
            const float s = redS[wv][tid];
            const int   c = redC[wv][tid];
            if (s > bs || (s == bs && c < bc)) { bs = s; bc = c; }
        }
        float* vor = d_out + (size_t)V * TOPK + C;
        float* msc = vor + V;
        vor[m0 + tid] = (float)bc;
        msc[m0 + tid] = bs;
    }
}

// ------------------------------------------------------------------- launcher
extern "C" void kernel_launch(void* const* d_in, const int* in_sizes, int n_in,
                              void* d_out, int out_size, void* d_ws, size_t ws_size,
                              hipStream_t stream) {
    const float* emb = (const float*)d_in[0];
    const int*   x   = (const int*)d_in[1];
    const int*   Kp  = (const int*)d_in[2];
    float* out = (float*)d_out;

    // workspace layout: norm_emb [V*D f32] | cmap_i [V*50 i32] | cand_i [1024 i32]
    float* nemb   = (float*)d_ws;
    int*   cmap_i = (int*)((char*)d_ws + (size_t)V * D * sizeof(float));
    int*   cand_i = (int*)((char*)cmap_i + (size_t)V * TOPK * sizeof(int));

    norm_kernel<<<V / 8, 256, 0, stream>>>(emb, nemb);
    cmap_kernel<<<MBLKS, 256, 0, stream>>>(nemb, out, cmap_i);
    cand_kernel<<<1, 1024, 0, stream>>>(x, Kp, cmap_i, out + (size_t)V * TOPK, cand_i);
    voronoi_kernel<<<MTILES, 256, 0, stream>>>(nemb, cand_i, Kp, out);
}